// VoxelizationWithMapping_58609123721982
// MI455X (gfx1250) — compile-verified
//
#include <hip/hip_runtime.h>
#include <stdint.h>

// ---------------- problem constants (match reference) ----------------
#define XD 200
#define YD 200
#define ZD 16
#define BD 8
#define XYZ (XD * YD * ZD)            // 640,000
#define NKEYS (BD * XYZ)              // 5,120,000 possible voxel keys
#define SCAN_TILE 4096                // flags per scan block (256 thr * 16)
#define NSCANBLK (NKEYS / SCAN_TILE)  // 1250 (exact)

// ---------------- kernel 0: clear flags (uint4 stores) ----------------
__global__ void vx_clear_flags(uint4* flags16, int n16) {
    int i = blockIdx.x * blockDim.x + threadIdx.x;
    if (i < n16) flags16[i] = make_uint4(0u, 0u, 0u, 0u);
}

// ---------------- kernel 1: pre-fill unq rows with -1 ----------------
__global__ void vx_fill_neg(int4* unq, int nrows) {
    int i = blockIdx.x * blockDim.x + threadIdx.x;
    if (i < nrows) unq[i] = make_int4(-1, -1, -1, -1);
}

// ---------------- kernel 2: quantize points -> keys, set flags -------
// Stages each lane's 16B point into LDS with the gfx1250 async
// global->LDS DMA path (ASYNCcnt), then computes the voxel key.
__global__ void vx_keys(const float4* __restrict__ pts, uint8_t* __restrict__ flags,
                        int* __restrict__ keyslot, int P, int N) {
    __shared__ float4 sh[256];
    int p = blockIdx.x * blockDim.x + threadIdx.x;
    if (p >= P) return;

    const float4* gp = pts + p;
    uint32_t lds_addr = (uint32_t)(uintptr_t)(&sh[threadIdx.x]);
    // async copy 16B global -> LDS (per lane), tracked by ASYNCcnt
    asm volatile("global_load_async_to_lds_b128 %0, %1, off"
                 :: "v"(lds_addr), "v"(gp) : "memory");
    asm volatile("s_wait_asynccnt 0" ::: "memory");
    float4 v = sh[threadIdx.x];

    // idx = trunc( (S * (v - lo)) / (hi - lo) )  -- same op order as reference
    float fx = (200.0f * (v.x + 40.0f)) / 80.0f;
    float fy = (200.0f * (v.y + 40.0f)) / 80.0f;
    float fz = (16.0f  * (v.z +  1.0f)) / 6.4f;
    int qx = (int)fx; int qy = (int)fy; int qz = (int)fz;
    // points are strictly in-range by construction; clamp only as OOB guard
    qx = min(max(qx, 0), XD - 1);
    qy = min(max(qy, 0), YD - 1);
    qz = min(max(qz, 0), ZD - 1);
    int b = p / N;

    int key = ((b * XD + qx) * YD + qy) * ZD + qz;
    flags[key] = (uint8_t)1;   // racy same-value stores: benign
    keyslot[p] = key;          // stash key in the unq_inv output slot
}

// ---------------- kernel 3: block-local exclusive scan of flags ------
__global__ void vx_scan1(const uint8_t* __restrict__ flags, uint32_t* __restrict__ ranks,
                         uint32_t* __restrict__ bsums) {
    __shared__ uint32_t s[256];
    const int t = threadIdx.x;
    const int base = blockIdx.x * SCAN_TILE + t * 16;

    uint4 w = *(const uint4*)(flags + base);
    uint32_t wd[4] = { w.x, w.y, w.z, w.w };
    uint32_t f[16];
    uint32_t sum = 0;
#pragma unroll
    for (int i = 0; i < 16; ++i) { f[i] = (wd[i >> 2] >> ((i & 3) * 8)) & 0xFFu; sum += f[i]; }

    s[t] = sum; __syncthreads();
#pragma unroll
    for (int off = 1; off < 256; off <<= 1) {
        uint32_t v = (t >= off) ? s[t - off] : 0u;
        __syncthreads();
        s[t] += v;
        __syncthreads();
    }
    uint32_t run = s[t] - sum;   // exclusive prefix of this thread within block

    uint32_t r[16];
#pragma unroll
    for (int i = 0; i < 16; ++i) { r[i] = run; run += f[i]; }
    uint4* rv = (uint4*)(ranks + base);
    rv[0] = make_uint4(r[0], r[1], r[2], r[3]);
    rv[1] = make_uint4(r[4], r[5], r[6], r[7]);
    rv[2] = make_uint4(r[8], r[9], r[10], r[11]);
    rv[3] = make_uint4(r[12], r[13], r[14], r[15]);

    if (t == 255) bsums[blockIdx.x] = s[255];
}

// ---------------- kernel 4: single-block exclusive scan of bsums -----
__global__ void vx_scan2(uint32_t* bsums, int n) {
    __shared__ uint32_t s[256];
    const int t = threadIdx.x;
    uint32_t running = 0;
    for (int chunk = 0; chunk < n; chunk += 256) {
        int i = chunk + t;
        uint32_t v = (i < n) ? bsums[i] : 0u;
        s[t] = v; __syncthreads();
#pragma unroll
        for (int off = 1; off < 256; off <<= 1) {
            uint32_t x = (t >= off) ? s[t - off] : 0u;
            __syncthreads();
            s[t] += x;
            __syncthreads();
        }
        uint32_t incl = s[t];
        uint32_t tot = s[255];
        __syncthreads();                 // all reads done before next chunk writes
        if (i < n) bsums[i] = running + incl - v;   // exclusive + carry
        running += tot;
    }
}

// ---------------- kernel 5: emit decoded unique voxel rows -----------
__global__ void vx_emit(const uint8_t* __restrict__ flags, const uint32_t* __restrict__ ranks,
                        const uint32_t* __restrict__ bsums, int4* __restrict__ unq) {
    int k = blockIdx.x * blockDim.x + threadIdx.x;
    if (k >= NKEYS) return;
    if (!flags[k]) return;
    uint32_t r = ranks[k] + bsums[k >> 12];   // k/4096 == k/SCAN_TILE
    int b = k / XYZ;
    int rem = k - b * XYZ;
    int x = rem / (YD * ZD);
    rem -= x * (YD * ZD);
    int y = rem / ZD;
    int z = rem - y * ZD;
    unq[r] = make_int4(b, x, y, z);
}

// ---------------- kernel 6: inverse map (key -> rank), in place ------
__global__ void vx_inv(const uint32_t* __restrict__ ranks, const uint32_t* __restrict__ bsums,
                       int* __restrict__ inv, int P) {
    int p = blockIdx.x * blockDim.x + threadIdx.x;
    if (p >= P) return;
    uint32_t k = (uint32_t)inv[p];
    inv[p] = (int)(ranks[k] + bsums[k >> 12]);
}

// ---------------- launcher ----------------
extern "C" void kernel_launch(void* const* d_in, const int* in_sizes, int n_in,
                              void* d_out, int out_size, void* d_ws, size_t ws_size,
                              hipStream_t stream) {
    const float4* pts = (const float4*)d_in[0];
    const int P = in_sizes[0] / 4;      // 4,000,000 points
    const int N = P / BD;               // 500,000 points per batch

    int* out = (int*)d_out;
    int4* unq = (int4*)out;             // P rows of 4 ints
    int* inv = out + 4 * (size_t)P;     // P ints

    // workspace layout: flags (u8, NKEYS) | ranks (u32, NKEYS) | bsums (u32, NSCANBLK)
    uint8_t* flags = (uint8_t*)d_ws;
    uint32_t* ranks = (uint32_t*)((char*)d_ws + NKEYS);
    uint32_t* bsums = (uint32_t*)((char*)d_ws + NKEYS + (size_t)NKEYS * 4);

    const int TB = 256;
    const int n16 = NKEYS / 16;

    vx_clear_flags<<<(n16 + TB - 1) / TB, TB, 0, stream>>>((uint4*)flags, n16);
    vx_fill_neg<<<(P + TB - 1) / TB, TB, 0, stream>>>(unq, P);
    vx_keys<<<(P + TB - 1) / TB, TB, 0, stream>>>(pts, flags, inv, P, N);
    vx_scan1<<<NSCANBLK, TB, 0, stream>>>(flags, ranks, bsums);
    vx_scan2<<<1, TB, 0, stream>>>(bsums, NSCANBLK);
    vx_emit<<<(NKEYS + TB - 1) / TB, TB, 0, stream>>>(flags, ranks, bsums, unq);
    vx_inv<<<(P + TB - 1) / TB, TB, 0, stream>>>(ranks, bsums, inv, P);
}